// QLSTMGen224_65481071405483
// MI455X (gfx1250) — compile-verified
//
#include <hip/hip_runtime.h>
#include <hip/hip_bf16.h>

// ---------------------------------------------------------------------------
// Quantum LSTM on MI455X (gfx1250). All state-vector circuits are factored
// into (per-sample product-state encoder) x (fixed 256x256 unitary), turning
// the simulation into fp32 complex GEMMs executed with V_WMMA_F32_16X16X4_F32.
// B planes are stored K-pair-interleaved so each lane's WMMA B-fragment is a
// single global_load_b64; each wave computes two 16x16 M-subtiles so every B
// fragment feeds 2 (feat) / 4 (gate) WMMAs.
// ---------------------------------------------------------------------------

typedef __attribute__((ext_vector_type(2))) float v2f;
typedef __attribute__((ext_vector_type(8))) float v8f;

#define NQ 8
#define DIM 256              // 2^NQ
#define BATCH 512
#define SEQ 64
#define NROWS (SEQ * BATCH)  // 32768
#define UPLANE (DIM * DIM)   // 65536 floats per plane
#define BM 32                // M rows per block

// ws layout (float units)
#define OFF_U 0                               // [5 circuits][2 planes][256][256]
#define OFF_FEATS (10 * UPLANE)               // [64][512][8]
#define OFF_ANGLES (OFF_FEATS + NROWS * NQ)   // [4][512][8]
#define OFF_Z (OFF_ANGLES + 4 * BATCH * NQ)   // [4][512][8]
#define OFF_HX (OFF_Z + 4 * BATCH * NQ)       // [512][8]
#define OFF_CX (OFF_HX + BATCH * NQ)          // [512][8]

// Pair-interleaved B plane addressing: element (k, col) of the logical
// [K=256][N=256] B matrix lives at  (k>>1)*512 + col*2 + (k&1).
__device__ __forceinline__ int bpk(int k, int col) {
  return (k >> 1) * (2 * DIM) + col * 2 + (k & 1);
}

// ---------------------------------------------------------------------------
// Kernel 1: build the 5 fixed unitaries by simulating each circuit on all 256
// basis states. Block = one basis column j (256 threads, state in LDS).
// Stores U^T (j = contraction row, t = output col) pair-interleaved.
// ---------------------------------------------------------------------------

__device__ __forceinline__ void apply_1q_lds(float* sre, float* sim_, int t, int w,
                                             float u00r, float u00i, float u01r, float u01i,
                                             float u10r, float u10i, float u11r, float u11i) {
  int m = 1 << (7 - w);
  int i0 = t & ~m, i1 = t | m;
  float a0r = sre[i0], a0i = sim_[i0];
  float a1r = sre[i1], a1i = sim_[i1];
  __syncthreads();
  float rr, ri;
  if (t & m) {
    rr = u10r * a0r - u10i * a0i + u11r * a1r - u11i * a1i;
    ri = u10r * a0i + u10i * a0r + u11r * a1i + u11i * a1r;
  } else {
    rr = u00r * a0r - u00i * a0i + u01r * a1r - u01i * a1i;
    ri = u00r * a0i + u00i * a0r + u01r * a1i + u01i * a1r;
  }
  sre[t] = rr; sim_[t] = ri;
  __syncthreads();
}

__device__ __forceinline__ void apply_rx_lds(float* sre, float* sim_, int t, int w, float th) {
  float c = cosf(th * 0.5f), s = sinf(th * 0.5f);
  // RX = [[c, -i s], [-i s, c]]
  apply_1q_lds(sre, sim_, t, w, c, 0.f, 0.f, -s, 0.f, -s, c, 0.f);
}

__device__ __forceinline__ void apply_ry_lds(float* sre, float* sim_, int t, int w, float th) {
  float c = cosf(th * 0.5f), s = sinf(th * 0.5f);
  // RY = [[c, -s], [s, c]]
  apply_1q_lds(sre, sim_, t, w, c, 0.f, -s, 0.f, s, 0.f, c, 0.f);
}

__device__ __forceinline__ void apply_cnot_lds(float* sre, float* sim_, int t, int cw, int tw) {
  int mc = 1 << (7 - cw), mt = 1 << (7 - tw);
  int src = (t & mc) ? (t ^ mt) : t;
  float r = sre[src], i = sim_[src];
  __syncthreads();
  sre[t] = r; sim_[t] = i;
  __syncthreads();
}

__global__ void k_build_unitaries(const float* __restrict__ var_params,
                                  const float* __restrict__ theta_rx,
                                  const float* __restrict__ theta_ry,
                                  const float* __restrict__ th_f,
                                  const float* __restrict__ th_i,
                                  const float* __restrict__ th_u,
                                  const float* __restrict__ th_o,
                                  float* __restrict__ ws) {
  __shared__ float sre[DIM];
  __shared__ float sim_[DIM];
  int t = threadIdx.x;   // amplitude index 0..255
  int j = blockIdx.x;    // basis column 0..255
  int c = blockIdx.y;    // circuit: 0=feat, 1..4=f,i,u,o
  sre[t] = (t == j) ? 1.f : 0.f;
  sim_[t] = 0.f;
  __syncthreads();

  if (c == 0) {
    for (int jj = 0; jj < 20; ++jj) {
      int w = jj % NQ;
      float th = var_params[jj];
      if ((jj & 1) == 0) apply_rx_lds(sre, sim_, t, w, th);
      else               apply_ry_lds(sre, sim_, t, w, th);
      apply_cnot_lds(sre, sim_, t, w, (w + 1) % NQ);
    }
    float trx = theta_rx[0], try_ = theta_ry[0];
    for (int w = 0; w < NQ; ++w) {
      apply_rx_lds(sre, sim_, t, w, trx);
      apply_ry_lds(sre, sim_, t, w, try_);
    }
  } else {
    const float* th = (c == 1) ? th_f : (c == 2) ? th_i : (c == 3) ? th_u : th_o;
    for (int w = 0; w < NQ; ++w) apply_rx_lds(sre, sim_, t, w, th[w]);
    for (int w = 0; w < NQ - 1; ++w) apply_cnot_lds(sre, sim_, t, w, w + 1);
  }

  float* Ur = ws + (size_t)(c * 2 + 0) * UPLANE;
  float* Ui = ws + (size_t)(c * 2 + 1) * UPLANE;
  // pair-interleaved: (j, t) -> (j>>1)*512 + t*2 + (j&1)
  int o = bpk(j, t);
  Ur[o] = sre[t];
  Ui[o] = sim_[t];
}

// ---------------------------------------------------------------------------
// Kernel 2: feature extractor for ALL timesteps at once.
// rows m = t*512+b (32768). A[m][j] = prod-state from RY(x) (real).
// psi = A * U_feat^T (complex B), feats = |psi|^2 @ signs.
// Block: 512 threads = 16 waves; wave owns one 16-col N-tile and BOTH 16-row
// M-subtiles. K = 256.
// ---------------------------------------------------------------------------
__global__ void k_feat_gemm(const float* __restrict__ inputs,
                            const float* __restrict__ ws_u,
                            float* __restrict__ feats) {
  __shared__ __align__(16) float Alds[BM * DIM];  // [32 rows][256] : A then probs
  __shared__ float cs[BM][NQ][2];

  int tid = threadIdx.x;
  int M0 = blockIdx.x * BM;

  // phase 1: cos/sin of half-angles per (row, wire)
  if (tid < BM * NQ) {
    int r = tid >> 3, w = tid & 7;
    float x = inputs[(M0 + r) * NQ + w];
    cs[r][w][0] = cosf(x * 0.5f);
    cs[r][w][1] = sinf(x * 0.5f);
  }
  __syncthreads();

  // phase 2: product state A[r][j] (real)
  for (int idx = tid; idx < BM * DIM; idx += 512) {
    int r = idx >> 8, k = idx & 255;
    float v = 1.f;
#pragma unroll
    for (int w = 0; w < NQ; ++w) {
      int bit = (k >> (7 - w)) & 1;
      v *= cs[r][w][bit];
    }
    Alds[idx] = v;
  }
  __syncthreads();

  // phase 3: WMMA GEMM (Cr = A*Ur, Ci = A*Ui) for 2 M-subtiles
  int wave = tid >> 5;
  int lane = tid & 31;
  int n0 = wave * 16;
  int mrow = lane & 15;          // M row within tile (A) / N col (B)
  int khalf = (lane >> 4) * 2;   // K sub-offset 0 or 2

  const float* Br = ws_u + 0 * UPLANE;
  const float* Bi = ws_u + 1 * UPLANE;
  int col = n0 + mrow;

  v8f Cr0 = {}, Ci0 = {}, Cr1 = {}, Ci1 = {};
  for (int k0 = 0; k0 < DIM; k0 += 4) {
    v2f a0 = *(const v2f*)&Alds[mrow * DIM + k0 + khalf];
    v2f a1 = *(const v2f*)&Alds[(16 + mrow) * DIM + k0 + khalf];
    int boff = bpk(k0 + khalf, col);   // even k -> aligned b64 pair
    v2f br = *(const v2f*)&Br[boff];
    v2f bi = *(const v2f*)&Bi[boff];
    Cr0 = __builtin_amdgcn_wmma_f32_16x16x4_f32(false, a0, false, br, (short)0, Cr0, false, false);
    Ci0 = __builtin_amdgcn_wmma_f32_16x16x4_f32(false, a0, false, bi, (short)0, Ci0, false, false);
    Cr1 = __builtin_amdgcn_wmma_f32_16x16x4_f32(false, a1, false, br, (short)0, Cr1, false, false);
    Ci1 = __builtin_amdgcn_wmma_f32_16x16x4_f32(false, a1, false, bi, (short)0, Ci1, false, false);
  }
  __syncthreads();  // done reading A; reuse Alds for probs

  // phase 4: probs -> LDS. C/D layout: lane<16 -> rows 0..7, lane>=16 -> 8..15
  int rbase = (lane >> 4) * 8;
  int kcol = n0 + (lane & 15);
#pragma unroll
  for (int v = 0; v < 8; ++v) {
    float p0 = Cr0[v] * Cr0[v] + Ci0[v] * Ci0[v];
    float p1 = Cr1[v] * Cr1[v] + Ci1[v] * Ci1[v];
    Alds[(rbase + v) * DIM + kcol] = p0;
    Alds[(16 + rbase + v) * DIM + kcol] = p1;
  }
  __syncthreads();

  // phase 5: <Z_w> reduction
  if (tid < BM * NQ) {
    int r = tid >> 3, w = tid & 7;
    int sh = 7 - w;
    float acc = 0.f;
    for (int k = 0; k < DIM; ++k) {
      float s = 1.f - 2.f * (float)((k >> sh) & 1);
      acc += s * Alds[r * DIM + k];
    }
    feats[(M0 + r) * NQ + w] = acc;
  }
}

// ---------------------------------------------------------------------------
// Kernel 3a: angles[g][b][w] = comb[b] . W_g[w] + b_g[w], comb = [feats_t, hx]
// ---------------------------------------------------------------------------
__global__ void k_angles(const float* __restrict__ feats_t, const float* __restrict__ hx,
                         const float* __restrict__ W0, const float* __restrict__ B0,
                         const float* __restrict__ W1, const float* __restrict__ B1,
                         const float* __restrict__ W2, const float* __restrict__ B2,
                         const float* __restrict__ W3, const float* __restrict__ B3,
                         float* __restrict__ angles) {
  int idx = blockIdx.x * blockDim.x + threadIdx.x;  // 0..16383
  int w = idx & 7;
  int b = (idx >> 3) & 511;
  int g = idx >> 12;
  const float* W = (g == 0) ? W0 : (g == 1) ? W1 : (g == 2) ? W2 : W3;
  const float* Bb = (g == 0) ? B0 : (g == 1) ? B1 : (g == 2) ? B2 : B3;
  float acc = Bb[w];
#pragma unroll
  for (int j = 0; j < NQ; ++j) acc += feats_t[b * NQ + j] * W[w * 16 + j];
#pragma unroll
  for (int j = 0; j < NQ; ++j) acc += hx[b * NQ + j] * W[w * 16 + 8 + j];
  angles[idx] = acc;
}

// ---------------------------------------------------------------------------
// Kernel 3b: quantum gate GEMM (complex psi0 x complex U_g), 4 gates in
// gridDim.y. psi0[j] = r(j) * (-i)^popcount(j) from RX encoding.
// Negation is carried in an LDS plane Ain = -Ai so the complex GEMM is all
// adds: Cr = Ar*Br + Ain*Bi ; Ci = Ar*Bi + Ai*Br.
// ---------------------------------------------------------------------------
__global__ void k_gate_gemm(const float* __restrict__ angles,
                            const float* __restrict__ ws_u,
                            float* __restrict__ zbuf) {
  __shared__ __align__(16) float Ar[BM * DIM];   // reused for probs at the end
  __shared__ __align__(16) float Ai[BM * DIM];
  __shared__ __align__(16) float Ain[BM * DIM];
  __shared__ float cs[BM][NQ][2];

  int tid = threadIdx.x;
  int M0 = blockIdx.x * BM;   // batch rows
  int g = blockIdx.y;         // gate 0..3

  if (tid < BM * NQ) {
    int r = tid >> 3, w = tid & 7;
    float a = angles[g * (BATCH * NQ) + (M0 + r) * NQ + w];
    cs[r][w][0] = cosf(a * 0.5f);
    cs[r][w][1] = sinf(a * 0.5f);
  }
  __syncthreads();

  for (int idx = tid; idx < BM * DIM; idx += 512) {
    int r = idx >> 8, k = idx & 255;
    float v = 1.f;
#pragma unroll
    for (int w = 0; w < NQ; ++w) {
      int bit = (k >> (7 - w)) & 1;
      v *= cs[r][w][bit];
    }
    int p = __popc(k) & 3;  // (-i)^popcount phase
    float ar = 0.f, ai = 0.f;
    if (p == 0) ar = v;
    else if (p == 1) ai = -v;
    else if (p == 2) ar = -v;
    else ai = v;
    Ar[idx] = ar;
    Ai[idx] = ai;
    Ain[idx] = -ai;
  }
  __syncthreads();

  int wave = tid >> 5;
  int lane = tid & 31;
  int n0 = wave * 16;
  int mrow = lane & 15;
  int khalf = (lane >> 4) * 2;

  const float* Br = ws_u + (size_t)((g + 1) * 2 + 0) * UPLANE;
  const float* Bi = ws_u + (size_t)((g + 1) * 2 + 1) * UPLANE;
  int col = n0 + mrow;

  v8f Cr0 = {}, Ci0 = {}, Cr1 = {}, Ci1 = {};
  for (int k0 = 0; k0 < DIM; k0 += 4) {
    int aoff0 = mrow * DIM + k0 + khalf;
    int aoff1 = (16 + mrow) * DIM + k0 + khalf;
    v2f ar0 = *(const v2f*)&Ar[aoff0];
    v2f ai0 = *(const v2f*)&Ai[aoff0];
    v2f an0 = *(const v2f*)&Ain[aoff0];
    v2f ar1 = *(const v2f*)&Ar[aoff1];
    v2f ai1 = *(const v2f*)&Ai[aoff1];
    v2f an1 = *(const v2f*)&Ain[aoff1];
    int boff = bpk(k0 + khalf, col);
    v2f br = *(const v2f*)&Br[boff];
    v2f bi = *(const v2f*)&Bi[boff];
    Cr0 = __builtin_amdgcn_wmma_f32_16x16x4_f32(false, ar0, false, br, (short)0, Cr0, false, false);
    Cr0 = __builtin_amdgcn_wmma_f32_16x16x4_f32(false, an0, false, bi, (short)0, Cr0, false, false);
    Ci0 = __builtin_amdgcn_wmma_f32_16x16x4_f32(false, ar0, false, bi, (short)0, Ci0, false, false);
    Ci0 = __builtin_amdgcn_wmma_f32_16x16x4_f32(false, ai0, false, br, (short)0, Ci0, false, false);
    Cr1 = __builtin_amdgcn_wmma_f32_16x16x4_f32(false, ar1, false, br, (short)0, Cr1, false, false);
    Cr1 = __builtin_amdgcn_wmma_f32_16x16x4_f32(false, an1, false, bi, (short)0, Cr1, false, false);
    Ci1 = __builtin_amdgcn_wmma_f32_16x16x4_f32(false, ar1, false, bi, (short)0, Ci1, false, false);
    Ci1 = __builtin_amdgcn_wmma_f32_16x16x4_f32(false, ai1, false, br, (short)0, Ci1, false, false);
  }
  __syncthreads();

  int rbase = (lane >> 4) * 8;
  int kcol = n0 + (lane & 15);
#pragma unroll
  for (int v = 0; v < 8; ++v) {
    float p0 = Cr0[v] * Cr0[v] + Ci0[v] * Ci0[v];
    float p1 = Cr1[v] * Cr1[v] + Ci1[v] * Ci1[v];
    Ar[(rbase + v) * DIM + kcol] = p0;
    Ar[(16 + rbase + v) * DIM + kcol] = p1;
  }
  __syncthreads();

  if (tid < BM * NQ) {
    int r = tid >> 3, w = tid & 7;
    int sh = 7 - w;
    float acc = 0.f;
    for (int k = 0; k < DIM; ++k) {
      float s = 1.f - 2.f * (float)((k >> sh) & 1);
      acc += s * Ar[r * DIM + k];
    }
    zbuf[g * (BATCH * NQ) + (M0 + r) * NQ + w] = acc;
  }
}

// ---------------------------------------------------------------------------
// Kernel 3c: LSTM cell update + tagger head for one timestep.
// Block 256 = 32 batch x 8 wires; grid 16.
// ---------------------------------------------------------------------------
__global__ void k_lstm(const float* __restrict__ zbuf,
                       float* __restrict__ hx, float* __restrict__ cx,
                       const float* __restrict__ W_head, const float* __restrict__ b_head,
                       float* __restrict__ out_t) {
  __shared__ float hs[256];
  int tid = threadIdx.x;
  int b = blockIdx.x * 32 + (tid >> 3);
  int w = tid & 7;
  int i = b * NQ + w;
  float zf = zbuf[0 * (BATCH * NQ) + i];
  float zi = zbuf[1 * (BATCH * NQ) + i];
  float zu = zbuf[2 * (BATCH * NQ) + i];
  float zo = zbuf[3 * (BATCH * NQ) + i];
  float f = 1.f / (1.f + expf(-zf));
  float ig = 1.f / (1.f + expf(-zi));
  float g = tanhf(zu);
  float o = 1.f / (1.f + expf(-zo));
  float c = f * cx[i] + ig * g;
  float h = o * tanhf(c);
  cx[i] = c;
  hx[i] = h;
  hs[tid] = h * W_head[w];
  __syncthreads();
  if (w == 0) {
    float acc = b_head[0];
#pragma unroll
    for (int j = 0; j < 8; ++j) acc += hs[tid + j];
    out_t[b] = acc;
  }
}

// ---------------------------------------------------------------------------
extern "C" void kernel_launch(void* const* d_in, const int* in_sizes, int n_in,
                              void* d_out, int out_size, void* d_ws, size_t ws_size,
                              hipStream_t stream) {
  (void)in_sizes; (void)n_in; (void)out_size; (void)ws_size;
  // setup_inputs dict order:
  const float* inputs    = (const float*)d_in[0];
  const float* W[4]      = {(const float*)d_in[1], (const float*)d_in[4],
                            (const float*)d_in[7], (const float*)d_in[10]};
  const float* bv[4]     = {(const float*)d_in[2], (const float*)d_in[5],
                            (const float*)d_in[8], (const float*)d_in[11]};
  const float* th[4]     = {(const float*)d_in[3], (const float*)d_in[6],
                            (const float*)d_in[9], (const float*)d_in[12]};
  const float* var_params = (const float*)d_in[13];
  const float* theta_rx   = (const float*)d_in[14];
  const float* theta_ry   = (const float*)d_in[15];
  const float* W_head     = (const float*)d_in[16];
  const float* b_head     = (const float*)d_in[17];

  float* ws      = (float*)d_ws;
  float* out     = (float*)d_out;
  float* Uplanes = ws + OFF_U;
  float* feats   = ws + OFF_FEATS;
  float* angles  = ws + OFF_ANGLES;
  float* zbuf    = ws + OFF_Z;
  float* hx      = ws + OFF_HX;
  float* cx      = ws + OFF_CX;

  hipMemsetAsync(hx, 0, BATCH * NQ * sizeof(float), stream);
  hipMemsetAsync(cx, 0, BATCH * NQ * sizeof(float), stream);

  // 1) fixed-circuit unitaries (5 circuits x 256 basis columns)
  k_build_unitaries<<<dim3(256, 5), 256, 0, stream>>>(
      var_params, theta_rx, theta_ry, th[0], th[1], th[2], th[3], Uplanes);

  // 2) feature extractor for all 64 timesteps at once (fully parallel)
  k_feat_gemm<<<NROWS / BM, 512, 0, stream>>>(inputs, Uplanes, feats);

  // 3) sequential recurrence
  for (int t = 0; t < SEQ; ++t) {
    k_angles<<<64, 256, 0, stream>>>(feats + (size_t)t * BATCH * NQ, hx,
                                     W[0], bv[0], W[1], bv[1], W[2], bv[2], W[3], bv[3],
                                     angles);
    k_gate_gemm<<<dim3(BATCH / BM, 4), 512, 0, stream>>>(angles, Uplanes, zbuf);
    k_lstm<<<16, 256, 0, stream>>>(zbuf, hx, cx, W_head, b_head, out + (size_t)t * BATCH);
  }
}